// AttentionHead_27625229648265
// MI455X (gfx1250) — compile-verified
//
#include <hip/hip_runtime.h>
#include <math.h>

typedef __attribute__((ext_vector_type(2))) float v2f;
typedef __attribute__((ext_vector_type(8))) float v8f;

#define BATCH 16
#define SEQ   2048
#define DIM   64
#define SCALE 0.125f  // 1/sqrt(64)

// ---------------------------------------------------------------------------
// Kernel 1: QKV projection.  q/k/v[b,s,:] = x[b,s,:] @ W + b
// grid.x = B*S/16 tiles of 16 rows; 4 waves per block, wave w owns out dims
// [16w, 16w+16).  K=64 -> 16 steps of V_WMMA_F32_16X16X4_F32 per projection.
// ---------------------------------------------------------------------------
__global__ __launch_bounds__(128) void qkv_proj_kernel(
    const float* __restrict__ x,
    const float* __restrict__ Wq, const float* __restrict__ bq,
    const float* __restrict__ Wk, const float* __restrict__ bk,
    const float* __restrict__ Wv, const float* __restrict__ bv,
    float* __restrict__ qo, float* __restrict__ ko, float* __restrict__ vo)
{
    const int row_base = blockIdx.x * 16;          // flattened (b*S + s) tile
    const int wave = threadIdx.x >> 5;             // 0..3 -> 16-wide d slice
    const int lane = threadIdx.x & 31;
    const int l16  = lane & 15;                    // A: row M ; B/C: col N
    const int hi   = lane >> 4;
    const int k0   = hi * 2;                       // K sub-offset per half
    const int ncol = (wave << 4) + l16;

    // A fragments: x tile (16 rows x 64), row M = l16, K-step kk covers
    // dims {4kk+k0, 4kk+k0+1} for this half-wave.
    v2f a[16];
    const float* xrow = x + (size_t)(row_base + l16) * DIM;
#pragma unroll
    for (int kk = 0; kk < 16; ++kk)
        a[kk] = *(const v2f*)(xrow + kk * 4 + k0);

    const float* W[3]   = {Wq, Wk, Wv};
    const float* bias[3] = {bq, bk, bv};
    float*       outp[3] = {qo, ko, vo};

#pragma unroll
    for (int m = 0; m < 3; ++m) {
        v8f c = {};
#pragma unroll
        for (int kk = 0; kk < 16; ++kk) {
            const float* wp = W[m] + (size_t)(kk * 4 + k0) * DIM + ncol;
            v2f b;
            b.x = wp[0];        // W[4kk+k0  ][ncol]
            b.y = wp[DIM];      // W[4kk+k0+1][ncol]
            c = __builtin_amdgcn_wmma_f32_16x16x4_f32(
                    false, a[kk], false, b, (short)0, c, false, false);
        }
        const float bb = bias[m][ncol];
        float* op = outp[m] + (size_t)row_base * DIM + ncol;
#pragma unroll
        for (int r = 0; r < 8; ++r)
            op[(size_t)(r + hi * 8) * DIM] = c[r] + bb;   // row M = r + 8*hi
    }
}

// ---------------------------------------------------------------------------
// Kernel 2: causal flash attention, one wave per (batch, 16-query tile).
// Streams 16-key blocks: S = QK^T (16 fp32 WMMAs), online softmax with
// shfl_xor row reductions (wave32, 16-lane row groups), P transposed
// C-layout -> A-layout via padded LDS, then O += P@V (16 fp32 WMMAs).
// Scores matrix never hits memory.
// ---------------------------------------------------------------------------
__global__ __launch_bounds__(32) void flash_attn_kernel(
    const float* __restrict__ q, const float* __restrict__ k,
    const float* __restrict__ v, float* __restrict__ out)
{
    __shared__ float p_lds[16 * 18];   // stride 18: conflict-free, float2-aligned

    const int b    = blockIdx.y;
    const int qb   = blockIdx.x;               // query tile 0..127
    const int lane = threadIdx.x & 31;
    const int l16  = lane & 15;
    const int hi   = lane >> 4;
    const int k0   = hi * 2;
    const int q_base = qb * 16;

    const float* qp = q + ((size_t)b * SEQ + q_base) * DIM;
    const float* kp = k + (size_t)b * SEQ * DIM;
    const float* vp = v + (size_t)b * SEQ * DIM;

    // Q tile in A layout, resident in VGPRs for the whole pass.
    v2f qa[16];
#pragma unroll
    for (int kk = 0; kk < 16; ++kk)
        qa[kk] = *(const v2f*)(qp + (size_t)l16 * DIM + kk * 4 + k0);

    v8f o0 = {}, o1 = {}, o2 = {}, o3 = {};
    float mrow[8], lrow[8];
#pragma unroll
    for (int r = 0; r < 8; ++r) { mrow[r] = -INFINITY; lrow[r] = 0.0f; }

    for (int jb = 0; jb <= qb; ++jb) {
        const int kbase = jb * 16;

        // ---- scores: S = Q @ K^T ; B[d][n] = K[kbase+n][d] ----
        v8f sc = {};
#pragma unroll
        for (int kk = 0; kk < 16; ++kk) {
            v2f bf = *(const v2f*)(kp + (size_t)(kbase + l16) * DIM + kk * 4 + k0);
            sc = __builtin_amdgcn_wmma_f32_16x16x4_f32(
                     false, qa[kk], false, bf, (short)0, sc, false, false);
        }

        // ---- scale, causal mask, online softmax ----
        const int kcol = kbase + l16;
        float pr[8], alpha[8];
#pragma unroll
        for (int r = 0; r < 8; ++r) {
            float s = sc[r] * SCALE;
            const int qrow = q_base + r + hi * 8;
            if (kcol > qrow) s = -INFINITY;
            pr[r] = s;
        }
#pragma unroll
        for (int r = 0; r < 8; ++r) {
            float rm = pr[r];                       // row max over 16 lanes
            rm = fmaxf(rm, __shfl_xor(rm, 1, 16));
            rm = fmaxf(rm, __shfl_xor(rm, 2, 16));
            rm = fmaxf(rm, __shfl_xor(rm, 4, 16));
            rm = fmaxf(rm, __shfl_xor(rm, 8, 16));
            const float mn = fmaxf(mrow[r], rm);
            alpha[r] = __expf(mrow[r] - mn);
            mrow[r]  = mn;
            const float p = __expf(pr[r] - mn);     // masked -> exp(-inf)=0
            pr[r] = p;
            float rs = p;                           // row sum over 16 lanes
            rs += __shfl_xor(rs, 1, 16);
            rs += __shfl_xor(rs, 2, 16);
            rs += __shfl_xor(rs, 4, 16);
            rs += __shfl_xor(rs, 8, 16);
            lrow[r] = lrow[r] * alpha[r] + rs;
        }
#pragma unroll
        for (int r = 0; r < 8; ++r) {
            o0[r] *= alpha[r]; o1[r] *= alpha[r];
            o2[r] *= alpha[r]; o3[r] *= alpha[r];
        }

        // ---- P: C layout -> LDS -> A layout ----
        __syncthreads();
#pragma unroll
        for (int r = 0; r < 8; ++r)
            p_lds[(r + hi * 8) * 18 + l16] = pr[r];
        __syncthreads();

        // ---- O += P @ V ; K=16 -> 4 WMMA steps, 4 d-tiles ----
#pragma unroll
        for (int kk2 = 0; kk2 < 4; ++kk2) {
            const int col = kk2 * 4 + k0;
            v2f pa = *(const v2f*)(&p_lds[l16 * 18 + col]);
            const float* vr0 = vp + (size_t)(kbase + col) * DIM;
            const float* vr1 = vr0 + DIM;
            v2f bf;
            bf.x = vr0[l16];      bf.y = vr1[l16];
            o0 = __builtin_amdgcn_wmma_f32_16x16x4_f32(false, pa, false, bf, (short)0, o0, false, false);
            bf.x = vr0[16 + l16]; bf.y = vr1[16 + l16];
            o1 = __builtin_amdgcn_wmma_f32_16x16x4_f32(false, pa, false, bf, (short)0, o1, false, false);
            bf.x = vr0[32 + l16]; bf.y = vr1[32 + l16];
            o2 = __builtin_amdgcn_wmma_f32_16x16x4_f32(false, pa, false, bf, (short)0, o2, false, false);
            bf.x = vr0[48 + l16]; bf.y = vr1[48 + l16];
            o3 = __builtin_amdgcn_wmma_f32_16x16x4_f32(false, pa, false, bf, (short)0, o3, false, false);
        }
    }

    // ---- normalize by l and store ----
    float inv[8];
#pragma unroll
    for (int r = 0; r < 8; ++r) inv[r] = 1.0f / lrow[r];
    float* op = out + ((size_t)b * SEQ + q_base) * DIM;
#pragma unroll
    for (int r = 0; r < 8; ++r) {
        const size_t row = (size_t)(r + hi * 8) * DIM;
        op[row +  0 + l16] = o0[r] * inv[r];
        op[row + 16 + l16] = o1[r] * inv[r];
        op[row + 32 + l16] = o2[r] * inv[r];
        op[row + 48 + l16] = o3[r] * inv[r];
    }
}

// ---------------------------------------------------------------------------
extern "C" void kernel_launch(void* const* d_in, const int* in_sizes, int n_in,
                              void* d_out, int out_size, void* d_ws, size_t ws_size,
                              hipStream_t stream) {
    const float* x  = (const float*)d_in[0];
    const float* Wq = (const float*)d_in[1];
    const float* bq = (const float*)d_in[2];
    const float* Wk = (const float*)d_in[3];
    const float* bk = (const float*)d_in[4];
    const float* Wv = (const float*)d_in[5];
    const float* bv = (const float*)d_in[6];
    float* out = (float*)d_out;

    // Workspace: q | k | v, each B*S*D fp32 (8 MB) -> 24 MB total.
    const size_t bsd = (size_t)BATCH * SEQ * DIM;
    float* qw = (float*)d_ws;
    float* kw = qw + bsd;
    float* vw = kw + bsd;

    const int tiles = (BATCH * SEQ) / 16;   // 2048
    qkv_proj_kernel<<<tiles, 128, 0, stream>>>(x, Wq, bq, Wk, bk, Wv, bv,
                                               qw, kw, vw);
    flash_attn_kernel<<<dim3(SEQ / 16, BATCH), 32, 0, stream>>>(qw, kw, vw, out);
}